// Sobelxy_39427799777291
// MI455X (gfx1250) — compile-verified
//
#include <hip/hip_runtime.h>
#include <stdint.h>

// Sobel |gx|+|gy| on [16,64,256,256] fp32.
// Memory-bound: ~537MB traffic -> ~23us floor at 23.3 TB/s HBM.
// Strategy: async global->LDS tile staging (CDNA5 path), separable stencil,
// b128 everywhere, fully coalesced 256B-per-halfwave stores.

#define W 256
#define H 256
#define TILE_ROWS 16
#define LDS_ROWS (TILE_ROWS + 2)
#define LDS_STRIDE 260            // floats per LDS row (256 + 4 pad -> bank spread)
#define THREADS 256

__global__ __launch_bounds__(THREADS)
void sobel_tile_kernel(const float* __restrict__ in, float* __restrict__ out) {
    __shared__ __align__(16) float lds[LDS_ROWS * LDS_STRIDE];

    const int strip = blockIdx.x;          // 0..H/TILE_ROWS-1
    const int plane = blockIdx.y;          // 0..B*C-1
    const int tid   = threadIdx.x;
    const int y0    = strip * TILE_ROWS;

    const float* pin  = in  + (size_t)plane * (W * H);
    float*       pout = out + (size_t)plane * (W * H);

    // ---------- Stage 1: fill 18x256 input tile into LDS ----------
    // 18 rows * 64 chunks of 16B = 1152 chunks, strided over 256 threads.
    // In-range rows: CDNA5 async copy global->LDS (no VGPR staging).
    // Out-of-range halo rows (zero padding): ds-store zeros.
    for (int idx = tid; idx < LDS_ROWS * (W / 4); idx += THREADS) {
        const int row = idx >> 6;          // 0..17
        const int cg  = idx & 63;          // 16-byte chunk within row
        const int gy  = y0 - 1 + row;
        float* ldst = &lds[row * LDS_STRIDE + cg * 4];
        if (gy >= 0 && gy < H) {
            const float* gsrc = pin + (size_t)gy * W + cg * 4;
            const uint32_t lds_off = (uint32_t)(uintptr_t)ldst;   // low 32b of generic = LDS byte offset
            const uint64_t gaddr   = (uint64_t)(uintptr_t)gsrc;
            asm volatile("global_load_async_to_lds_b128 %0, %1, off"
                         :
                         : "v"(lds_off), "v"(gaddr)
                         : "memory");
        } else {
            *(float4*)ldst = make_float4(0.f, 0.f, 0.f, 0.f);
        }
    }
    asm volatile("s_wait_asynccnt 0x0" ::: "memory");
    __syncthreads();

    // ---------- Stage 2: separable Sobel from LDS ----------
    // Thread -> output row r = tid>>4; 4 segments of 4 cols at c = 64q + 4*(tid&15).
    // Store pattern: 16 consecutive lanes write 256 contiguous bytes per instr.
    const int r  = tid >> 4;               // 0..15 (tile row)
    const int gg = tid & 15;               // 0..15 (segment lane)

    float* prow = pout + (size_t)(y0 + r) * W;

    #pragma unroll
    for (int q = 0; q < 4; ++q) {
        const int c = q * 64 + gg * 4;     // first output column of this segment

        // Load 3 input rows, columns c-1 .. c+4 (6 each).
        float a0[6], a1[6], a2[6];
        #pragma unroll
        for (int k = 0; k < 3; ++k) {
            const float* base = &lds[(r + k) * LDS_STRIDE + c];
            float4 m = *(const float4*)base;
            float lft = (c == 0)       ? 0.f : base[-1];   // left zero pad
            float rgt = (c == W - 4)   ? 0.f : base[4];    // right zero pad
            float* a = (k == 0) ? a0 : (k == 1) ? a1 : a2;
            a[0] = lft; a[1] = m.x; a[2] = m.y; a[3] = m.z; a[4] = m.w; a[5] = rgt;
        }

        // Vertical pass: s = smooth [1,2,1], d = diff [1,0,-1] (top - bottom).
        float s[6], d[6];
        #pragma unroll
        for (int j = 0; j < 6; ++j) {
            s[j] = fmaf(2.f, a1[j], a0[j] + a2[j]);
            d[j] = a0[j] - a2[j];
        }

        // Horizontal pass: gx = s[x+1]-s[x-1]; gy = d[x-1]+2d[x]+d[x+1].
        float4 o;
        {
            float gx0 = s[2] - s[0];
            float gx1 = s[3] - s[1];
            float gx2 = s[4] - s[2];
            float gx3 = s[5] - s[3];
            float gy0 = fmaf(2.f, d[1], d[0] + d[2]);
            float gy1 = fmaf(2.f, d[2], d[1] + d[3]);
            float gy2 = fmaf(2.f, d[3], d[2] + d[4]);
            float gy3 = fmaf(2.f, d[4], d[3] + d[5]);
            o.x = fabsf(gx0) + fabsf(gy0);                 // folds to v_add_f32 |a|,|b|
            o.y = fabsf(gx1) + fabsf(gy1);
            o.z = fabsf(gx2) + fabsf(gy2);
            o.w = fabsf(gx3) + fabsf(gy3);
        }
        *(float4*)(prow + c) = o;                          // 16B aligned, coalesced
    }
}

extern "C" void kernel_launch(void* const* d_in, const int* in_sizes, int n_in,
                              void* d_out, int out_size, void* d_ws, size_t ws_size,
                              hipStream_t stream) {
    (void)n_in; (void)out_size; (void)d_ws; (void)ws_size;
    const float* x   = (const float*)d_in[0];
    float*       out = (float*)d_out;

    const int n      = in_sizes[0];        // 16*64*256*256
    const int planes = n / (W * H);        // 1024

    dim3 grid(H / TILE_ROWS, planes, 1);   // (16, 1024)
    sobel_tile_kernel<<<grid, THREADS, 0, stream>>>(x, out);
}